// EntityModel_crf_7078106104208
// MI455X (gfx1250) — compile-verified
//
#include <hip/hip_runtime.h>
#include <hip/hip_bf16.h>

// Problem constants from the reference
#define B_ 64
#define S_ 512
#define H_ 768
#define T_ 32

typedef float v2f __attribute__((ext_vector_type(2)));
typedef float v8f __attribute__((ext_vector_type(8)));

#define BK 64
#define SA_STRIDE 68   // 16-row A tile, stride 68 floats -> WMMA A reads hit distinct banks

// ---------------------------------------------------------------------------
// Kernel 0: zero the scalar output (d_out is poisoned by the harness)
// ---------------------------------------------------------------------------
__global__ void zero_out_kernel(float* out) { out[0] = 0.0f; }

// ---------------------------------------------------------------------------
// Kernel 1: feat[b,s,t] = hidden[b,s,:] @ W_out[:,t] + b_out[t]
// fp32 WMMA 16x16x4. Block = 256 threads = 8 waves. Each wave computes a
// 16-row x 32-col tile of feat (two 16x16 accumulators). Grid = 32768/128.
//
// LDS layouts:
//   sA: per-wave 16x64 tile, row stride 68 -> A fragment reads (b64) are
//       bank-conflict free (stride 68 % 64 = 4, 16 lanes cover all banks).
//   sW: W chunk stored in WMMA B-fragment order:
//       sW[g*128 + col*4 + c*2 + j] = W[ko + 4g + 2c + j][col]
//       so each lane's B fragment is one aligned float2 (single ds_load_b64),
//       lanes 0-15 on banks {4k,4k+1}, lanes 16-31 on {4k+2,4k+3}: no conflict.
// ---------------------------------------------------------------------------
__global__ __launch_bounds__(256) void gemm_feat_wmma(
    const float* __restrict__ hidden,   // [B*S, H]
    const float* __restrict__ W,        // [H, T]
    const float* __restrict__ bias,     // [T]
    float* __restrict__ feat)           // [B*S, T]
{
    __shared__ float sA[8 * 16 * SA_STRIDE];  // per-wave 16x64 A tiles (34 KB)
    __shared__ float sW[(BK / 4) * 128];      // 16 groups * 128 floats (8 KB)

    const int tid  = threadIdx.x;
    const int w    = tid >> 5;        // wave id 0..7
    const int lane = tid & 31;
    const int l16  = lane & 15;
    const int hi   = lane >> 4;       // 0: lanes 0-15, 1: lanes 16-31
    const int m0   = blockIdx.x * 128 + w * 16;   // first row of this wave's strip

    v8f acc0 = {};   // cols 0..15
    v8f acc1 = {};   // cols 16..31

    const float* aBase = hidden + (size_t)m0 * H_;

    for (int ko = 0; ko < H_; ko += BK) {
        __syncthreads();   // protect LDS from previous iteration's readers

        // Stage W chunk (64x32) into B-fragment order, coalesced global reads.
        #pragma unroll
        for (int j = 0; j < 8; ++j) {
            int idx = tid + 256 * j;          // 0..2047
            int r = idx >> 5, col = idx & 31; // r: row within chunk
            int g  = r >> 2;                  // kk-group
            int c  = (r >> 1) & 1;            // which half-K (lanes 0-15 vs 16-31)
            int jj = r & 1;                   // v0 / v1 within fragment
            sW[g * 128 + col * 4 + c * 2 + jj] = W[(ko + r) * T_ + col];
        }
        // Stage this wave's A tile: 16 rows x 64 cols as float4 (coalesced 512B/instr)
        #pragma unroll
        for (int i = 0; i < 8; ++i) {
            int f  = i * 32 + lane;           // float4 index 0..255
            int r  = f >> 4, c4 = f & 15;
            float4 v = *(const float4*)(aBase + (size_t)r * H_ + ko + c4 * 4);
            *(float4*)(&sA[(w * 16 + r) * SA_STRIDE + c4 * 4]) = v;
        }
        __syncthreads();

        const float* sAw    = &sA[(w * 16 + l16) * SA_STRIDE] + 2 * hi;
        const float* sWlane = &sW[l16 * 4 + hi * 2];
        #pragma unroll
        for (int kk = 0; kk < BK; kk += 4) {
            const int g = kk >> 2;
            // A fragment: A[l16][kk + 2*hi .. +1]  (ISA 32-bit A layout)
            v2f a  = *(const v2f*)(sAw + kk);
            // B fragments: one b64 each, two N-tiles 256B apart (fusable to 2addr)
            v2f b0 = *(const v2f*)(sWlane + g * 128);
            v2f b1 = *(const v2f*)(sWlane + g * 128 + 64);
            acc0 = __builtin_amdgcn_wmma_f32_16x16x4_f32(false, a, false, b0,
                                                         (short)0, acc0, false, false);
            acc1 = __builtin_amdgcn_wmma_f32_16x16x4_f32(false, a, false, b1,
                                                         (short)0, acc1, false, false);
        }
    }

    // Epilogue: C layout (VGPR j: lanes 0-15 -> row j, lanes 16-31 -> row j+8)
    const float bias0 = bias[l16];
    const float bias1 = bias[16 + l16];
    float* outBase = feat + (size_t)m0 * T_;
    #pragma unroll
    for (int j = 0; j < 8; ++j) {
        int row = j + 8 * hi;
        outBase[row * T_ + l16]      = acc0[j] + bias0;
        outBase[row * T_ + 16 + l16] = acc1[j] + bias1;
    }
}

// ---------------------------------------------------------------------------
// Kernel 2: CRF forward + gold score. One block (32 waves) per batch element.
// wave w = destination state t', lane l = source state t.
// ---------------------------------------------------------------------------
__global__ __launch_bounds__(1024) void crf_forward(
    const float* __restrict__ feat,     // [B, S, T]
    const int*  __restrict__ mask,      // [B, S]
    const int*  __restrict__ tgt,       // [B, S]
    const float* __restrict__ trans,    // [T, T]
    float* __restrict__ out)
{
    const int b   = blockIdx.x;
    const int tid = threadIdx.x;
    const int w   = tid >> 5;   // t'
    const int l   = tid & 31;   // t

    __shared__ float s_alpha[2][T_];
    __shared__ float s_d[T_];
    __shared__ float s_gold;
    __shared__ int   s_len;

    const float my_trans = trans[l * T_ + w];   // register-resident for the whole scan
    const float* featb = feat + (size_t)b * S_ * T_;
    const int*   maskb = mask + b * S_;
    const int*   tgtb  = tgt  + b * S_;

    if (w == 0) {
        // sequence length and alpha0
        int s = 0;
        for (int k = l; k < S_; k += 32) s += maskb[k];
        #pragma unroll
        for (int off = 16; off; off >>= 1) s += __shfl_xor(s, off, 32);
        if (l == 0) s_len = s - 1;
        s_alpha[0][l] = featb[l];
    }
    if (w == 1) {
        // gold score
        float g = 0.0f;
        for (int i = l; i < S_; i += 32) {
            int t = tgtb[i];
            float e = featb[i * T_ + t];
            if (i > 0) e += trans[tgtb[i - 1] * T_ + t];
            if (maskb[i]) g += e;
        }
        #pragma unroll
        for (int off = 16; off; off >>= 1) g += __shfl_xor(g, off, 32);
        if (l == 0) s_gold = g;
    }
    __syncthreads();
    const int len = s_len;
    if (w == 0 && len == 0) s_d[l] = s_alpha[0][l];
    __syncthreads();

    int cur = 0;
    for (int i = 1; i < S_; ++i) {
        float a = s_alpha[cur][l] + my_trans;
        float m = a;
        #pragma unroll
        for (int off = 16; off; off >>= 1) m = fmaxf(m, __shfl_xor(m, off, 32));
        float e = __expf(a - m);
        #pragma unroll
        for (int off = 16; off; off >>= 1) e += __shfl_xor(e, off, 32);
        float na = m + __logf(e) + featb[i * T_ + w];
        int nxt = cur ^ 1;
        if (l == 0) {
            s_alpha[nxt][w] = na;
            if (i == len) s_d[w] = na;
        }
        cur = nxt;
        __syncthreads();
    }

    if (w == 0) {
        float d = s_d[l];
        float m = d;
        #pragma unroll
        for (int off = 16; off; off >>= 1) m = fmaxf(m, __shfl_xor(m, off, 32));
        float e = __expf(d - m);
        #pragma unroll
        for (int off = 16; off; off >>= 1) e += __shfl_xor(e, off, 32);
        if (l == 0) atomicAdd(out, (m + __logf(e)) - s_gold);
    }
}

// ---------------------------------------------------------------------------
extern "C" void kernel_launch(void* const* d_in, const int* in_sizes, int n_in,
                              void* d_out, int out_size, void* d_ws, size_t ws_size,
                              hipStream_t stream) {
    const float* hidden = (const float*)d_in[0];   // [B,S,H] f32
    const int*   mask   = (const int*)  d_in[1];   // [B,S]   i32
    const int*   tgt    = (const int*)  d_in[2];   // [B,S]   i32 (jax default int)
    const float* W      = (const float*)d_in[3];   // [H,T]   f32
    const float* bias   = (const float*)d_in[4];   // [T]     f32
    const float* trans  = (const float*)d_in[5];   // [T,T]   f32
    float* out  = (float*)d_out;
    float* feat = (float*)d_ws;                    // [B*S, T] = 4 MB scratch

    zero_out_kernel<<<1, 1, 0, stream>>>(out);
    gemm_feat_wmma<<<(B_ * S_) / 128, 256, 0, stream>>>(hidden, W, bias, feat);
    crf_forward<<<B_, 1024, 0, stream>>>(feat, mask, tgt, trans, out);
}